// MCALayer_32040456028196
// MI455X (gfx1250) — compile-verified
//
#include <hip/hip_runtime.h>
#include <cmath>
#include <cstdint>

#define B_   32
#define C_   256
#define H_   64
#define W_   64
#define HID_ 16

typedef float v2f __attribute__((ext_vector_type(2)));
typedef float v8f __attribute__((ext_vector_type(8)));

// LDS plane layout with zero halo so every stencil tap is an unconditional load.
// addr(p,c,w) = 4 + p*PLANEF + (c+1)*SXS + w,  c in [-1,256], w in [-1,64].
// Bank for 16 consecutive c at fixed w: (4c + w + 8) & 63 -> conflict-free.
// +4 guard keeps interior float4 fills 16B-aligned (b128 transfers).
#define SXS    68
#define PLANEF (259 * SXS)             // 17612 floats per plane (incl. slack)
#define OFF_SX   0
#define OFF_S2H  (4 + 3 * PLANEF)      // 52840
#define OFF_S2W  (OFF_S2H + C_)
#define OFF_RED  (OFF_S2W + C_)
#define OFF_RED2 (OFF_RED + 256)
#define OFF_VW   (OFF_RED2 + 256)
#define OFF_VARH (OFF_VW + W_)
#define SMEM_FLOATS (OFF_VARH + 8)     // 53936 floats = ~216 KB < 320 KB WGP LDS

__device__ __forceinline__ int sxidx(int p, int c, int w) {
    return 4 + p * PLANEF + (c + 1) * SXS + w;
}

// ---------------- kernel 1: var over (c,h) slices -> var_w[b,w] ----------------
__global__ void mca_varw(const float* __restrict__ x, float* __restrict__ varw) {
    __shared__ float r1[256];
    __shared__ float r2[256];
    const int b = blockIdx.x >> 6;     // / W_
    const int w = blockIdx.x & 63;
    const int tid = threadIdx.x;
    float s = 0.f, sq = 0.f;
    for (int i = tid; i < C_ * H_; i += 256) {
        int c = i >> 6, h = i & 63;
        float v = x[(((size_t)b * C_ + c) * H_ + h) * W_ + w];
        s += v; sq += v * v;
    }
    r1[tid] = s; r2[tid] = sq;
    __syncthreads();
    for (int st = 128; st > 0; st >>= 1) {
        if (tid < st) { r1[tid] += r1[tid + st]; r2[tid] += r2[tid + st]; }
        __syncthreads();
    }
    if (tid == 0) {
        const float N = (float)(C_ * H_);
        varw[blockIdx.x] = (r2[0] - r1[0] * r1[0] / N) / (N - 1.0f);
    }
}

// ---------------- branchless stencils (halo already zero-filled) ----------------
__device__ __forceinline__ float conv_h(const float* sx, const float* whr, int c, int w) {
    float a = 0.f;
#pragma unroll
    for (int dc = -1; dc <= 1; ++dc) {
        const float* r = &sx[sxidx(1, c + dc, w - 1)];
        a += whr[(dc + 1) * 3 + 0] * r[0]
           + whr[(dc + 1) * 3 + 1] * r[1]
           + whr[(dc + 1) * 3 + 2] * r[2];
    }
    return a;
}
// fused pair of xh stencils at (c, w) and (c, w+1): 3x4 window, 12 loads
__device__ __forceinline__ v2f conv_h2(const float* sx, const float* whr, int c, int w) {
    float ax = 0.f, ay = 0.f;
#pragma unroll
    for (int dc = -1; dc <= 1; ++dc) {
        const float* r = &sx[sxidx(1, c + dc, w - 1)];
        const float l0 = r[0], l1 = r[1], l2 = r[2], l3 = r[3];
        const float k0 = whr[(dc + 1) * 3 + 0];
        const float k1 = whr[(dc + 1) * 3 + 1];
        const float k2 = whr[(dc + 1) * 3 + 2];
        ax += k0 * l0 + k1 * l1 + k2 * l2;
        ay += k0 * l1 + k1 * l2 + k2 * l3;
    }
    v2f o; o.x = ax; o.y = ay; return o;
}
// conv over (h,c) plane at fixed w: xw stencil (kernel row -> h, col -> c)
__device__ __forceinline__ float conv_w(const float* sx, const float* wwr, int c, int w) {
    float a = 0.f;
#pragma unroll
    for (int dh = -1; dh <= 1; ++dh)
#pragma unroll
        for (int dc = -1; dc <= 1; ++dc)
            a += wwr[(dh + 1) * 3 + (dc + 1)] * sx[sxidx(1 + dh, c + dc, w)];
    return a;
}

// ---------------- kernel 2: fused conv + WMMA attention + softmax + blend ----------------
__global__ void __launch_bounds__(256, 1)
mca_main(const float* __restrict__ x,
         const float* __restrict__ wh, const float* __restrict__ bh,
         const float* __restrict__ ww, const float* __restrict__ bw,
         const float* __restrict__ fc_w, const float* __restrict__ fc_b,
         const float* __restrict__ fc2_w, const float* __restrict__ fc2_b,
         const float* __restrict__ varw, float* __restrict__ out) {
    extern __shared__ float smem[];
    float* sx    = smem + OFF_SX;
    float* s2h   = smem + OFF_S2H;
    float* s2w   = smem + OFF_S2W;
    float* red   = smem + OFF_RED;
    float* red2  = smem + OFF_RED2;
    float* vwl   = smem + OFF_VW;
    float* varhp = smem + OFF_VARH;

    const int tid = threadIdx.x;
    const int b = blockIdx.x >> 6;     // / H_
    const int h = blockIdx.x & 63;

    float whr[9], wwr[9];
#pragma unroll
    for (int k = 0; k < 9; ++k) { whr[k] = wh[k]; wwr[k] = ww[k]; }
    const float bhv = bh[0], bwv = bw[0], fc2bv = fc2_b[0];

    if (tid < W_) vwl[tid] = varw[b * W_ + tid];

    // ---- async DMA the 3 input planes (h-1,h,h+1) into LDS; ASYNCcnt-tracked,
    //      bypasses VGPRs and overlaps with the halo zeroing below ----
    for (int p = 0; p < 3; ++p) {
        const int hh = h - 1 + p;
        const bool ok = (hh >= 0) && (hh < H_);
        if (ok) {
            const float* src = x + (((size_t)b * C_) * H_ + hh) * W_;
            for (int i = tid; i < C_ * (W_ / 4); i += 256) {
                const int c  = i >> 4;
                const int w4 = (i & 15) << 2;
                // flat->LDS mapping truncates to addr[31:0], so low bits of the
                // generic pointer are the LDS byte address.
                unsigned lds_addr = (unsigned)(uintptr_t)(&sx[sxidx(p, c, w4)]);
                uint64_t gaddr = (uint64_t)(uintptr_t)(src + (size_t)c * H_ * W_ + w4);
                asm volatile("global_load_async_to_lds_b128 %0, %1, off"
                             :: "v"(lds_addr), "v"(gaddr) : "memory");
            }
        } else {
            for (int i = tid; i < C_ * (W_ / 4); i += 256) {
                const int c  = i >> 4;
                const int w4 = (i & 15) << 2;
                *(float4*)(&sx[sxidx(p, c, w4)]) = make_float4(0.f, 0.f, 0.f, 0.f);
            }
        }
    }

    // ---- zero the halo cells (disjoint from interior DMA targets) ----
    for (int i = tid; i < 3 * 258; i += 256) {        // columns w=-1, w=64
        const int p = i / 258;
        const int c = (i % 258) - 1;
        sx[sxidx(p, c, -1)] = 0.f;
        sx[sxidx(p, c, 64)] = 0.f;
    }
    for (int i = tid; i < 3 * 2 * 64; i += 256) {     // halo rows c=-1, c=256
        const int p = i / 128;
        const int j = i % 128;
        const int c = (j < 64) ? -1 : C_;
        sx[sxidx(p, c, j & 63)] = 0.f;
    }

    asm volatile("s_wait_asynccnt 0" ::: "memory");   // our async DMAs complete
    __syncthreads();                                  // visible to all waves

    // ---- var_h[b,h] from plane 1 (unbiased, N = C*W) ----
    {
        float s = 0.f, sq = 0.f;
        for (int i = tid; i < C_ * W_; i += 256) {
            const int c = i >> 6, w = i & 63;
            const float v = sx[sxidx(1, c, w)];
            s += v; sq += v * v;
        }
        red[tid] = s; red2[tid] = sq;
        __syncthreads();
        for (int st = 128; st > 0; st >>= 1) {
            if (tid < st) { red[tid] += red[tid + st]; red2[tid] += red2[tid + st]; }
            __syncthreads();
        }
        if (tid == 0) {
            const float N = (float)(C_ * W_);
            varhp[0] = (red2[0] - red[0] * red[0] / N) / (N - 1.0f);
        }
        __syncthreads();
    }
    const float var_h = varhp[0];

    // ---- attention logits via V_WMMA_F32_16X16X4_F32 ----
    // A 16x4 f32: lanes 0-15 hold K%4 in {0,1}, lanes 16-31 hold K%4 in {2,3}; M = lane&15.
    const int lane  = tid & 31;
    const int wv    = tid >> 5;
    const int n     = lane & 15;            // A row in tile / D column (hid index)
    const int khalf = (lane < 16) ? 0 : 2;

    v2f bfr[16];                            // B fragments (fc_w), loop-invariant
#pragma unroll
    for (int kk = 0; kk < 16; ++kk) {
        const int k0 = 4 * kk + khalf;
        bfr[kk].x = fc_w[k0 * HID_ + n];
        bfr[kk].y = fc_w[(k0 + 1) * HID_ + n];
    }
    const float fcbv = fc_b[n];
    const float f2v  = fc2_w[n];

    for (int t = 0; t < 2; ++t) {
        const int c0 = 16 * (wv + 8 * t);

        // branch H
        v8f acc = {0.f, 0.f, 0.f, 0.f, 0.f, 0.f, 0.f, 0.f};
#pragma unroll
        for (int kk = 0; kk < 16; ++kk) {
            const int k0 = 4 * kk + khalf;
            v2f cv = conv_h2(sx, whr, c0 + n, k0);
            v2f a;
            a.x = var_h * cv.x + bhv;
            a.y = var_h * cv.y + bhv;
            acc = __builtin_amdgcn_wmma_f32_16x16x4_f32(false, a, false, bfr[kk],
                                                        (short)0, acc, false, false);
        }
#pragma unroll
        for (int r = 0; r < 8; ++r) {
            float s1 = acc[r] + fcbv; s1 = s1 > 0.f ? s1 : 0.f;
            float p = f2v * s1;
            p += __shfl_xor(p, 8, 16);
            p += __shfl_xor(p, 4, 16);
            p += __shfl_xor(p, 2, 16);
            p += __shfl_xor(p, 1, 16);
            if (n == 0) s2h[c0 + r + ((lane >= 16) ? 8 : 0)] = p + fc2bv;
        }

        // branch W (var_w depends on K = w)
        v8f accw = {0.f, 0.f, 0.f, 0.f, 0.f, 0.f, 0.f, 0.f};
#pragma unroll
        for (int kk = 0; kk < 16; ++kk) {
            const int k0 = 4 * kk + khalf;
            v2f a;
            a.x = vwl[k0]     * conv_w(sx, wwr, c0 + n, k0)     + bwv;
            a.y = vwl[k0 + 1] * conv_w(sx, wwr, c0 + n, k0 + 1) + bwv;
            accw = __builtin_amdgcn_wmma_f32_16x16x4_f32(false, a, false, bfr[kk],
                                                         (short)0, accw, false, false);
        }
#pragma unroll
        for (int r = 0; r < 8; ++r) {
            float s1 = accw[r] + fcbv; s1 = s1 > 0.f ? s1 : 0.f;
            float p = f2v * s1;
            p += __shfl_xor(p, 8, 16);
            p += __shfl_xor(p, 4, 16);
            p += __shfl_xor(p, 2, 16);
            p += __shfl_xor(p, 1, 16);
            if (n == 0) s2w[c0 + r + ((lane >= 16) ? 8 : 0)] = p + fc2bv;
        }
    }
    __syncthreads();

    // ---- softmax over c (dim=1) for both branches ----
    const float vh  = s2h[tid];
    const float vw2 = s2w[tid];
    red[tid] = vh; red2[tid] = vw2;
    __syncthreads();
    for (int st = 128; st > 0; st >>= 1) {
        if (tid < st) {
            red[tid]  = fmaxf(red[tid],  red[tid + st]);
            red2[tid] = fmaxf(red2[tid], red2[tid + st]);
        }
        __syncthreads();
    }
    const float mh = red[0], mw = red2[0];
    __syncthreads();
    const float eh = expf(vh - mh), ew = expf(vw2 - mw);
    red[tid] = eh; red2[tid] = ew;
    __syncthreads();
    for (int st = 128; st > 0; st >>= 1) {
        if (tid < st) { red[tid] += red[tid + st]; red2[tid] += red2[tid + st]; }
        __syncthreads();
    }
    const float sh = red[0], sw = red2[0];
    __syncthreads();
    s2h[tid] = eh / sh;    // alpha_h[b, tid, h]
    s2w[tid] = ew / sw;    // alpha_w[b, tid, h]
    __syncthreads();

    // ---- final blend: out = 0.5*(xh*alpha_h + xw*alpha_w) ----
    for (int i = tid; i < C_ * W_; i += 256) {
        const int c = i >> 6, w = i & 63;
        const float xhv = var_h * conv_h(sx, whr, c, w) + bhv;
        const float xwv = vwl[w] * conv_w(sx, wwr, c, w) + bwv;
        out[(((size_t)b * C_ + c) * H_ + h) * W_ + w] = 0.5f * (xhv * s2h[c] + xwv * s2w[c]);
    }
}

extern "C" void kernel_launch(void* const* d_in, const int* in_sizes, int n_in,
                              void* d_out, int out_size, void* d_ws, size_t ws_size,
                              hipStream_t stream) {
    const float* x     = (const float*)d_in[0];
    const float* wh    = (const float*)d_in[1];
    const float* bh    = (const float*)d_in[2];
    const float* ww    = (const float*)d_in[3];
    const float* bw    = (const float*)d_in[4];
    const float* fc_w  = (const float*)d_in[5];
    const float* fc_b  = (const float*)d_in[6];
    const float* fc2_w = (const float*)d_in[7];
    const float* fc2_b = (const float*)d_in[8];
    float* varw = (float*)d_ws;                         // B_*W_ = 2048 floats
    float* out  = (float*)d_out;

    mca_varw<<<B_ * W_, 256, 0, stream>>>(x, varw);
    mca_main<<<B_ * H_, 256, SMEM_FLOATS * sizeof(float), stream>>>(
        x, wh, bh, ww, bw, fc_w, fc_b, fc2_w, fc2_b, varw, out);
}